// MaskedMatrixFactorization_72370198938200
// MI455X (gfx1250) — compile-verified
//
#include <hip/hip_runtime.h>
#include <hip/hip_bf16.h>

// ---------------------------------------------------------------------------
// MaskedMatrixFactorization for MI455X (gfx1250, wave32)
//
// Output layout (flat, return order):
//   [0]                : S_hat * interaction_mask   (U*I floats, ~800 MB, sparse)
//   [U*I]              : user_factor masked         (U*K)
//   [U*I+U*K]          : user_bias masked           (U)
//   [U*I+U*K+U]        : item_factor masked         (I*K)
//   [U*I+U*K+U+I*K]    : item_bias masked           (I)
//
// Cost model: writing 808 MB @ 23.3 TB/s ~= 35 us dominates. Compute is
// 4096 length-64 f32 dot products (~0.5 MFLOP) done with chained
// V_WMMA_F32_16X16X4_F32 (16 pairs per wave, diagonal of 16x16 D).
// ---------------------------------------------------------------------------

typedef __attribute__((ext_vector_type(2))) float v2f;
typedef __attribute__((ext_vector_type(4))) float v4f;
typedef __attribute__((ext_vector_type(8))) float v8f;

// --- Kernel 1: streaming zero-fill of the whole output (bandwidth bound) ---
__global__ void mmf_zero_fill(float* __restrict__ out, long long n) {
    long long idx    = (long long)blockIdx.x * blockDim.x + threadIdx.x;
    long long stride = (long long)gridDim.x * blockDim.x;
    long long n4     = n >> 2;
    v4f z = (v4f)0.0f;
    v4f* __restrict__ out4 = (v4f*)out;
    for (long long i = idx; i < n4; i += stride) {
        __builtin_nontemporal_store(z, out4 + i);   // global_store_b128, TH=NT
    }
    // scalar tail (out_size is divisible by 4 here, but be safe)
    long long tail = n & 3LL;
    if (idx < tail) out[n4 * 4 + idx] = 0.f;
}

// --- Kernel 2: masked row / bias copies (one block of K threads per pair) ---
__global__ void mmf_scatter_rows(const float* __restrict__ userW,
                                 const float* __restrict__ itemW,
                                 const float* __restrict__ userB,
                                 const float* __restrict__ itemB,
                                 const int*   __restrict__ uidx,
                                 const int*   __restrict__ iidx,
                                 float* __restrict__ outUF,
                                 float* __restrict__ outUB,
                                 float* __restrict__ outIF,
                                 float* __restrict__ outIB,
                                 int K) {
    int b = blockIdx.x;
    int t = threadIdx.x;
    int u = uidx[b];
    int i = iidx[b];
    if (t < K) {
        outUF[(long long)u * K + t] = userW[(long long)u * K + t];
        outIF[(long long)i * K + t] = itemW[(long long)i * K + t];
    }
    if (t == 0) {
        outUB[u] = userB[u];
        outIB[i] = itemB[i];
    }
}

// --- Kernel 3: interaction dot products via V_WMMA_F32_16X16X4_F32 ---------
// One wave handles 16 (u,i) pairs. A = 16 gathered user rows (16xK),
// B = 16 gathered item rows presented transposed (Kx16). D = A*B + C is
// 16x16; the diagonal D[p][p] = dot(user_p, item_p). K consumed 4 at a time.
//
// Fragment layout per ISA (32-bit A 16x4): lanes 0-15 hold M=lane with
// K+0/K+1 in the two VGPRs; lanes 16-31 hold M=lane-16 with K+2/K+3.
// B (4x16) mirrors this with N striped across lanes -> identical per-lane
// addressing into the gathered item row.
__global__ void mmf_interactions(const float* __restrict__ userW,
                                 const float* __restrict__ itemW,
                                 const float* __restrict__ userB,
                                 const float* __restrict__ itemB,
                                 const int*   __restrict__ uidx,
                                 const int*   __restrict__ iidx,
                                 float* __restrict__ outS,
                                 int I, int K, int B) {
    int wave = (int)((blockIdx.x * blockDim.x + threadIdx.x) >> 5);
    int lane = threadIdx.x & 31;
    int base = wave * 16;
    if (base >= B) return;                  // wave-uniform: EXEC stays all-1s

    int m    = lane & 15;                   // which pair (row of A / col of B)
    int koff = (lane >> 4) << 1;            // 0 for lanes 0-15, 2 for 16-31

    int u = uidx[base + m];
    int i = iidx[base + m];
    const float* __restrict__ arow = userW + (long long)u * K;
    const float* __restrict__ brow = itemW + (long long)i * K;

    v8f acc = {};
#pragma unroll
    for (int k = 0; k < 64; k += 4) {       // K == 64
        int c = k + koff;
        v2f a, b;
        a.x = arow[c]; a.y = arow[c + 1];
        b.x = brow[c]; b.y = brow[c + 1];
        // (neg_a, A, neg_b, B, c_mod, C, reuse_a, reuse_b)
        acc = __builtin_amdgcn_wmma_f32_16x16x4_f32(
            false, a, false, b, (short)0, acc, false, false);
    }

    // Diagonal of 16x16 f32 D: lanes 0-7 hold d[lane] in acc[lane];
    // lanes 24-31 hold d[lane-16] in acc[lane-24].
    bool have = (lane < 8) | (lane >= 24);
    int  r    = (lane < 8) ? lane : (lane - 24);
    float val = acc[0];
    if (r == 1) val = acc[1];
    if (r == 2) val = acc[2];
    if (r == 3) val = acc[3];
    if (r == 4) val = acc[4];
    if (r == 5) val = acc[5];
    if (r == 6) val = acc[6];
    if (r == 7) val = acc[7];

    if (have) {
        float s = val + userB[u] + itemB[i];
        outS[(long long)u * I + i] = s;
    }
}

extern "C" void kernel_launch(void* const* d_in, const int* in_sizes, int n_in,
                              void* d_out, int out_size, void* d_ws, size_t ws_size,
                              hipStream_t stream) {
    const float* userW = (const float*)d_in[0];   // (U, K)
    const float* itemW = (const float*)d_in[1];   // (I, K)
    const float* userB = (const float*)d_in[2];   // (U, 1)
    const float* itemB = (const float*)d_in[3];   // (I,)
    const int*   uidx  = (const int*)d_in[4];     // (B,)
    const int*   iidx  = (const int*)d_in[5];     // (B,)

    const int U = in_sizes[2];                    // user_bias has U elements
    const int I = in_sizes[3];                    // item_bias has I elements
    const int K = in_sizes[0] / U;                // 64
    const int B = in_sizes[4];                    // 4096

    float* out   = (float*)d_out;
    float* outS  = out;
    float* outUF = outS  + (long long)U * I;
    float* outUB = outUF + (long long)U * K;
    float* outIF = outUB + U;
    float* outIB = outIF + (long long)I * K;

    // 1) Zero the entire output (dominant, bandwidth-bound: ~808 MB of NT b128 stores)
    mmf_zero_fill<<<16384, 256, 0, stream>>>(out, (long long)out_size);

    // 2) Masked row / bias copies
    mmf_scatter_rows<<<B, (K < 64 ? 64 : K), 0, stream>>>(
        userW, itemW, userB, itemB, uidx, iidx, outUF, outUB, outIF, outIB, K);

    // 3) Interaction scatter with f32 WMMA (16 pairs per wave, 8 waves per block)
    int waves  = (B + 15) / 16;
    int blocks = (waves + 7) / 8;
    mmf_interactions<<<blocks, 256, 0, stream>>>(
        userW, itemW, userB, itemB, uidx, iidx, outS, I, K, B);
}